// CustomGate_549755814186
// MI455X (gfx1250) — compile-verified
//
#include <hip/hip_runtime.h>

// Gate: U = I_L (x) M (x) I_R on x in C^{8192 x 2048}, D=2, N=13, INDEX=5.
// L=32, R=128. Row pair for site p: i0=(p/128)*256 + p%128, i1=i0+128.
// Pure HBM-bandwidth-bound (AI ~ 1 flop/byte; floor ~11.5us at 23.3 TB/s).
// Strategy: 128-bit non-temporal loads/stores (in+out 268MB > 192MB L2, no
// reuse), global_prefetch_b8 at the exact +16MiB next-iteration stride.

typedef __attribute__((ext_vector_type(4))) float v4f;

#define DIMQ   8192
#define BATCHQ 2048
#define RW     128
#define PAIRS  (DIMQ / 2)          // 4096
#define KV     (BATCHQ / 4)        // 512 float4 column groups per row
#define CHUNKS (PAIRS * KV)        // 2,097,152 work items
#define NBLK   2048
#define NTHR   256
// stride = NBLK*NTHR = 524288 = 1024*KV  ->  p advances by 1024 per iter,
// i0 advances by 2048 rows, flat offset advances by 2048*2048 = 1<<22 floats.
#define PF_DELTA (1u << 22)

__global__ __launch_bounds__(NTHR) void qgate_apply(
    const float* __restrict__ Mr, const float* __restrict__ Mi,
    const float* __restrict__ xr, const float* __restrict__ xi,
    float* __restrict__ out)
{
    // 2x2 complex gate, uniform across the wave (row-major [a][b])
    const float m00r = Mr[0], m01r = Mr[1], m10r = Mr[2], m11r = Mr[3];
    const float m00i = Mi[0], m01i = Mi[1], m10i = Mi[2], m11i = Mi[3];

    const int stride = NBLK * NTHR;
    for (int t = blockIdx.x * NTHR + threadIdx.x; t < CHUNKS; t += stride) {
        const int kv = t & (KV - 1);          // float4 column group
        const int p  = t >> 9;                // pair/site index (t / KV)
        const int i0 = ((p >> 7) << 8) | (p & (RW - 1));
        const long off0 = (long)i0 * BATCHQ + (kv << 2);
        const long off1 = off0 + (long)RW * BATCHQ;

        // Speculative prefetch of next grid-stride iteration (+16MiB);
        // OOB translations are silently dropped (ISA 10.5).
        __builtin_prefetch(xr + off0 + PF_DELTA, 0, 0);
        __builtin_prefetch(xi + off0 + PF_DELTA, 0, 0);
        __builtin_prefetch(xr + off1 + PF_DELTA, 0, 0);
        __builtin_prefetch(xi + off1 + PF_DELTA, 0, 0);

        const v4f x0r = __builtin_nontemporal_load((const v4f*)(xr + off0));
        const v4f x1r = __builtin_nontemporal_load((const v4f*)(xr + off1));
        const v4f x0i = __builtin_nontemporal_load((const v4f*)(xi + off0));
        const v4f x1i = __builtin_nontemporal_load((const v4f*)(xi + off1));

        // y_a = M[a][0]*x0 + M[a][1]*x1  (complex)
        const v4f y0r = m00r * x0r - m00i * x0i + m01r * x1r - m01i * x1i;
        const v4f y0i = m00r * x0i + m00i * x0r + m01r * x1i + m01i * x1r;
        const v4f y1r = m10r * x0r - m10i * x0i + m11r * x1r - m11i * x1i;
        const v4f y1i = m10r * x0i + m10i * x0r + m11r * x1i + m11i * x1r;

        // complex64 layout: out[2*(i*B+k)] = re, +1 = im (interleaved)
        const v4f o0a = { y0r.x, y0i.x, y0r.y, y0i.y };
        const v4f o0b = { y0r.z, y0i.z, y0r.w, y0i.w };
        const v4f o1a = { y1r.x, y1i.x, y1r.y, y1i.y };
        const v4f o1b = { y1r.z, y1i.z, y1r.w, y1i.w };

        float* o0 = out + 2 * off0;
        float* o1 = out + 2 * off1;
        __builtin_nontemporal_store(o0a, (v4f*)(o0));
        __builtin_nontemporal_store(o0b, (v4f*)(o0 + 4));
        __builtin_nontemporal_store(o1a, (v4f*)(o1));
        __builtin_nontemporal_store(o1b, (v4f*)(o1 + 4));
    }
}

extern "C" void kernel_launch(void* const* d_in, const int* in_sizes, int n_in,
                              void* d_out, int out_size, void* d_ws, size_t ws_size,
                              hipStream_t stream) {
    (void)in_sizes; (void)n_in; (void)out_size; (void)d_ws; (void)ws_size;
    const float* Mr = (const float*)d_in[0];
    const float* Mi = (const float*)d_in[1];
    const float* xr = (const float*)d_in[2];
    const float* xi = (const float*)d_in[3];
    float* out = (float*)d_out;
    qgate_apply<<<dim3(NBLK), dim3(NTHR), 0, stream>>>(Mr, Mi, xr, xi, out);
}